// WeightedGCN_60129542734
// MI455X (gfx1250) — compile-verified
//
#include <hip/hip_runtime.h>
#include <hip/hip_bf16.h>

typedef __attribute__((ext_vector_type(2))) float v2f;
typedef __attribute__((ext_vector_type(8))) float v8f;

#define CH 64  // in/hid/out channels are all 64

// ---------------------------------------------------------------------------
// Degree / normalization kernels
// ---------------------------------------------------------------------------
__global__ void deg_init_kernel(float* __restrict__ deg, int n) {
    int i = blockIdx.x * blockDim.x + threadIdx.x;
    if (i < n) deg[i] = 1.0f;  // self-loop weight 1
}

__global__ void deg_accum_kernel(const int* __restrict__ col,
                                 const float* __restrict__ ew,
                                 float* __restrict__ deg, int nE) {
    int e = blockIdx.x * blockDim.x + threadIdx.x;
    if (e < nE) atomicAdd(&deg[col[e]], ew[e]);
}

__global__ void deg_to_dis_kernel(float* __restrict__ deg, int n) {
    int i = blockIdx.x * blockDim.x + threadIdx.x;
    if (i < n) {
        float d = deg[i];
        deg[i] = (d > 0.0f) ? rsqrtf(d) : 0.0f;
    }
}

// ---------------------------------------------------------------------------
// Dense GEMM  Y[M,64] = X[M,64] @ W[64,64]  via V_WMMA_F32_16X16X4_F32
// block = 128 threads = 4 waves; block handles one 16-row M tile,
// each wave handles one 16-col N tile. K=64 -> 16 wmma steps of K=4.
// ---------------------------------------------------------------------------
__global__ void __launch_bounds__(128)
gemm_wmma_f32_kernel(const float* __restrict__ X, const float* __restrict__ W,
                     float* __restrict__ Y, int M) {
    const int wave = threadIdx.x >> 5;          // 0..3 -> N tile
    const int lane = threadIdx.x & 31;
    const int half = lane >> 4;                 // 0 or 1
    const int lid  = lane & 15;
    const int mBase = blockIdx.x * 16;
    const int nBase = wave * 16;

    // clamp read row so WMMA runs with EXEC all-ones even on a ragged tail
    int mRead = mBase + lid;
    if (mRead >= M) mRead = M - 1;
    const float* xrow = X + (size_t)mRead * CH;

    v8f c = {};
#pragma unroll
    for (int kk = 0; kk < 16; ++kk) {
        const int k0 = kk * 4 + 2 * half;       // A: VGPR0 holds K=k0, VGPR1 K=k0+1
        v2f a, b;
        a.x = xrow[k0];
        a.y = xrow[k0 + 1];
        // B 4x16: VGPR r covers rows {r, r+2}; lane-half selects +2 offset
        b.x = W[(size_t)k0 * CH + nBase + lid];
        b.y = W[(size_t)(k0 + 1) * CH + nBase + lid];
        c = __builtin_amdgcn_wmma_f32_16x16x4_f32(
                /*neg_a=*/false, a, /*neg_b=*/false, b,
                /*c_mod=*/(short)0, c, /*reuse_a=*/false, /*reuse_b=*/false);
    }

#pragma unroll
    for (int r = 0; r < 8; ++r) {
        int m = mBase + r + 8 * half;           // C/D: VGPR r -> rows {r, r+8}
        if (m < M) Y[(size_t)m * CH + nBase + lid] = c[r];
    }
}

// ---------------------------------------------------------------------------
// Zero a float buffer (float4 stores)
// ---------------------------------------------------------------------------
__global__ void zero_kernel(float4* __restrict__ p, int n4) {
    int i = blockIdx.x * blockDim.x + threadIdx.x;
    if (i < n4) p[i] = make_float4(0.f, 0.f, 0.f, 0.f);
}

// ---------------------------------------------------------------------------
// Edge scatter: out[col] += dis[row]*ew*dis[col] * xw[row]
// One wave per edge, 2 channels per lane (float2), global f32 atomics.
// ---------------------------------------------------------------------------
__global__ void __launch_bounds__(256)
scatter_edges_kernel(const int* __restrict__ row, const int* __restrict__ col,
                     const float* __restrict__ ew, const float* __restrict__ dis,
                     const float* __restrict__ xw, float* __restrict__ out, int nE) {
    const int e = blockIdx.x * (blockDim.x >> 5) + (threadIdx.x >> 5);
    if (e >= nE) return;
    const int lane = threadIdx.x & 31;
    const int r = row[e];
    const int c = col[e];
    const float norm = dis[r] * ew[e] * dis[c];
    const float2 v = *(const float2*)(xw + (size_t)r * CH + 2 * lane);
    float* o = out + (size_t)c * CH + 2 * lane;
    atomicAdd(o,     norm * v.x);
    atomicAdd(o + 1, norm * v.y);
}

// ---------------------------------------------------------------------------
// Epilogue: out[i,ch] = (acc[i,ch] + dis[i]^2 * xw[i,ch] + bias[ch])  (+ReLU)
// Adds the self-loop contribution without atomics (runs after scatter).
// ---------------------------------------------------------------------------
__global__ void epilogue_kernel(float* __restrict__ acc, const float* __restrict__ xw,
                                const float* __restrict__ dis,
                                const float* __restrict__ bias,
                                int n, int relu) {
    int idx = blockIdx.x * blockDim.x + threadIdx.x;
    if (idx >= n * CH) return;
    const int i  = idx >> 6;
    const int ch = idx & (CH - 1);
    const float di = dis[i];
    float v = acc[idx] + di * di * xw[idx] + bias[ch];
    if (relu) v = fmaxf(v, 0.0f);
    acc[idx] = v;
}

// ---------------------------------------------------------------------------
extern "C" void kernel_launch(void* const* d_in, const int* in_sizes, int n_in,
                              void* d_out, int out_size, void* d_ws, size_t ws_size,
                              hipStream_t stream) {
    const float* x   = (const float*)d_in[0];
    const int*   ei  = (const int*)d_in[1];   // [2, E] int32 (JAX x64 disabled)
    const float* ew  = (const float*)d_in[2];
    const float* W1  = (const float*)d_in[3];
    const float* b1  = (const float*)d_in[4];
    const float* W2  = (const float*)d_in[5];
    const float* b2  = (const float*)d_in[6];
    float* out = (float*)d_out;

    const int N = in_sizes[0] / CH;   // 100000
    const int E = in_sizes[2];        // 1000000
    const int* row = ei;
    const int* col = ei + E;

    // workspace layout
    float* dis = (float*)d_ws;                 // N floats (deg -> dis in place)
    float* xw  = dis + 131072;                 // N*64 floats
    float* h   = xw + (size_t)N * CH;          // N*64 floats

    const int T = 256;
    const int nBlkN   = (N + T - 1) / T;
    const int nBlkE   = (E + T - 1) / T;
    const int nBlkNC  = (N * CH + T - 1) / T;
    const int nBlkNC4 = ((N * CH / 4) + T - 1) / T;
    const int edgeBlk = (E + 7) / 8;           // 8 waves (edges) per 256-thr block
    const int mTiles  = (N + 15) / 16;         // 6250

    // --- symmetric normalization coefficients ---
    deg_init_kernel<<<nBlkN, T, 0, stream>>>(dis, N);
    deg_accum_kernel<<<nBlkE, T, 0, stream>>>(col, ew, dis, E);
    deg_to_dis_kernel<<<nBlkN, T, 0, stream>>>(dis, N);

    // --- layer 1 ---
    gemm_wmma_f32_kernel<<<mTiles, 128, 0, stream>>>(x, W1, xw, N);
    zero_kernel<<<nBlkNC4, T, 0, stream>>>((float4*)h, N * CH / 4);
    scatter_edges_kernel<<<edgeBlk, T, 0, stream>>>(row, col, ew, dis, xw, h, E);
    epilogue_kernel<<<nBlkNC, T, 0, stream>>>(h, xw, dis, b1, N, /*relu=*/1);

    // --- layer 2 ---
    gemm_wmma_f32_kernel<<<mTiles, 128, 0, stream>>>(h, W2, xw, N);
    zero_kernel<<<nBlkNC4, T, 0, stream>>>((float4*)out, N * CH / 4);
    scatter_edges_kernel<<<edgeBlk, T, 0, stream>>>(row, col, ew, dis, xw, out, E);
    epilogue_kernel<<<nBlkNC, T, 0, stream>>>(out, xw, dis, b2, N, /*relu=*/0);
}